// HashedConv2d_7653631721654
// MI455X (gfx1250) — compile-verified
//
#include <hip/hip_runtime.h>
#include <hip/hip_bf16.h>

typedef float v2f __attribute__((ext_vector_type(2)));
typedef float v8f __attribute__((ext_vector_type(8)));

#define BUCKETS 256
#define CIN  64
#define COUT 64
#define HH   128
#define WW   128
#define BB   16
#define KPER 576           // CIN * 9
#define TILE_W 32          // pixels per workgroup
#define NCOL 34            // staged columns (32 + 2 halo)
#define CI_STRIDE 66       // padded ci-stride per column (64 + 2 pad; bank-conflict-free)

#if __has_builtin(__builtin_amdgcn_global_load_async_to_lds_b32)
#define HAVE_ASYNC_LDS 1
#endif

#if defined(HAVE_ASYNC_LDS)
typedef __attribute__((address_space(1))) int gas_int;
typedef __attribute__((address_space(3))) int las_int;

__device__ __forceinline__ void async_copy_f32(const float* g, float* l) {
    __builtin_amdgcn_global_load_async_to_lds_b32(
        (gas_int*)g, (las_int*)l, 0, 0);
}
__device__ __forceinline__ void async_wait_all() {
#if __has_builtin(__builtin_amdgcn_s_wait_asynccnt)
    __builtin_amdgcn_s_wait_asynccnt(0);
#else
    asm volatile("s_wait_asynccnt 0x0" ::: "memory");
#endif
}
#endif

// ---------------- Kernel 1: min/max over conv_weight ----------------
__global__ void minmax_kernel(const float* __restrict__ w, float* __restrict__ mm, int n) {
    __shared__ float smin[256];
    __shared__ float smax[256];
    int tid = threadIdx.x;
    float lo = 3.4e38f, hi = -3.4e38f;
    for (int i = tid; i < n; i += 256) {
        float v = w[i];
        lo = fminf(lo, v);
        hi = fmaxf(hi, v);
    }
    smin[tid] = lo; smax[tid] = hi;
    __syncthreads();
    for (int s = 128; s > 0; s >>= 1) {
        if (tid < s) {
            smin[tid] = fminf(smin[tid], smin[tid + s]);
            smax[tid] = fmaxf(smax[tid], smax[tid + s]);
        }
        __syncthreads();
    }
    if (tid == 0) { mm[0] = smin[0]; mm[1] = smax[0]; }
}

// ------- Kernel 2: bucketize + transpose both weight tensors to [co][pos][ci] -------
__global__ void hashw_kernel(const float* __restrict__ w, const float* __restrict__ hv,
                             const float* __restrict__ mm,
                             float* __restrict__ wt, float* __restrict__ hwt, int n) {
    int i = blockIdx.x * blockDim.x + threadIdx.x;
    if (i >= n) return;
    float lo = mm[0], hi = mm[1];
    float width = (hi - lo) * (1.0f / (float)BUCKETS);
    float wv = w[i];
    int idx = (int)floorf((wv - lo) / width);
    idx = min(max(idx, 0), BUCKETS - 1);
    // i = co*576 + ci*9 + pos  ->  j = (co*9 + pos)*64 + ci
    int co  = i / 576;
    int rem = i - co * 576;
    int ci  = rem / 9;
    int pos = rem - ci * 9;
    int j = (co * 9 + pos) * CIN + ci;
    wt[j]  = wv;
    hwt[j] = hv[idx];
}

// ---------------- Kernel 3: fused triple conv via f32 WMMA ----------------
// grid = (WW/TILE_W, HH, BB), block = 128 (4 waves; wave = cout tile of 16)
__global__ __launch_bounds__(128) void conv3_wmma_kernel(
    const float* __restrict__ x, const float* __restrict__ xh,
    const float* __restrict__ wt, const float* __restrict__ hwt,
    const float* __restrict__ bias, float* __restrict__ out)
{
    // layout: [kh(3)][col(34)][ci padded to 66]; b64 ci-pair loads, bank stride 2*m
    __shared__ float xs [3 * NCOL * CI_STRIDE];
    __shared__ float xhs[3 * NCOL * CI_STRIDE];

    const int tid  = threadIdx.x;
    const int lane = tid & 31;
    const int wave = tid >> 5;            // cout tile 0..3
    const int w0   = blockIdx.x * TILE_W; // pixel column base
    const int h    = blockIdx.y;
    const int b    = blockIdx.z;

    // ---- stage x / x_hash halo tiles: 3 rows x 34 cols x 64 ci ----
    const int E = CIN * 3 * NCOL;         // 6528 = 51 * 128 (no tail divergence)
    const size_t img = (size_t)b * CIN * HH * WW;
    for (int e = tid; e < E; e += 128) {
        int ci  = e / 102;                // consecutive tids -> consecutive cols (coalesced)
        int r   = e - ci * 102;
        int kh  = r / NCOL;
        int c   = r - kh * NCOL;
        int hin = h + kh - 1;
        int win = w0 + c - 1;
        bool ok = (hin >= 0) & (hin < HH) & (win >= 0) & (win < WW);
        size_t gi = img + ((size_t)ci * HH + (size_t)(ok ? hin : 0)) * WW + (size_t)(ok ? win : 0);
        int li = (kh * NCOL + c) * CI_STRIDE + ci;
#if defined(HAVE_ASYNC_LDS)
        if (ok) {
            async_copy_f32(x + gi,  &xs[li]);
            async_copy_f32(xh + gi, &xhs[li]);
        } else {
            xs[li] = 0.0f; xhs[li] = 0.0f;
        }
#else
        xs [li] = ok ? x [gi] : 0.0f;
        xhs[li] = ok ? xh[gi] : 0.0f;
#endif
    }

    const int m      = lane & 15;         // A row / B col within tile
    const int hi16   = lane >> 4;         // upper half-wave: K offset +2
    const int co_row = wave * 16 + m;     // cout this lane loads for A frags
    const float* wtb  = wt  + (size_t)co_row * KPER;  // [pos][ci] slice
    const float* hwtb = hwt + (size_t)co_row * KPER;

    __builtin_prefetch(wtb, 0, 3);
    __builtin_prefetch(hwtb, 0, 3);

#if defined(HAVE_ASYNC_LDS)
    async_wait_all();
#endif
    __syncthreads();

    v8f acc[3][2] = {};   // [out, hash_out, out_hash] x [2 N-tiles]

    #pragma unroll 1
    for (int pos = 0; pos < 9; ++pos) {   // (kh,kw) taps
        const int kh = pos / 3;
        const int kw = pos - kh * 3;
        const float* wp  = wtb  + pos * CIN;
        const float* hwp = hwtb + pos * CIN;
        // LDS column for this lane's pixel (nt=0); nt=1 is +16 columns
        const int col0 = (kh * NCOL + kw + m) * CI_STRIDE + hi16 * 2;
        const int col1 = col0 + 16 * CI_STRIDE;
        #pragma unroll 4
        for (int cc = 0; cc < 16; ++cc) { // K chunks of 4 channels
            const int ciA = cc * 4 + hi16 * 2;
            v2f aw = *(const v2f*)(wp  + ciA);   // global_load_b64
            v2f ah = *(const v2f*)(hwp + ciA);
            const int o = cc * 4;
            v2f bx0 = *(const v2f*)&xs [col0 + o];   // ds_load_b64
            v2f bh0 = *(const v2f*)&xhs[col0 + o];
            v2f bx1 = *(const v2f*)&xs [col1 + o];
            v2f bh1 = *(const v2f*)&xhs[col1 + o];
            acc[0][0] = __builtin_amdgcn_wmma_f32_16x16x4_f32(
                false, aw, false, bx0, (short)0, acc[0][0], false, false);
            acc[1][0] = __builtin_amdgcn_wmma_f32_16x16x4_f32(
                false, ah, false, bx0, (short)0, acc[1][0], false, false);
            acc[2][0] = __builtin_amdgcn_wmma_f32_16x16x4_f32(
                false, ah, false, bh0, (short)0, acc[2][0], false, false);
            acc[0][1] = __builtin_amdgcn_wmma_f32_16x16x4_f32(
                false, aw, false, bx1, (short)0, acc[0][1], false, false);
            acc[1][1] = __builtin_amdgcn_wmma_f32_16x16x4_f32(
                false, ah, false, bx1, (short)0, acc[1][1], false, false);
            acc[2][1] = __builtin_amdgcn_wmma_f32_16x16x4_f32(
                false, ah, false, bh1, (short)0, acc[2][1], false, false);
        }
    }

    // ---- epilogue: add bias, scatter D tiles (C layout: VGPR r -> M = r + 8*hi16) ----
    const size_t N1 = (size_t)BB * COUT * HH * WW;   // one output tensor
    #pragma unroll
    for (int nt = 0; nt < 2; ++nt) {
        const int col = w0 + nt * 16 + m;
        #pragma unroll
        for (int r = 0; r < 8; ++r) {
            const int co = wave * 16 + r + hi16 * 8;
            const float bv = bias[co];
            const size_t off = ((size_t)b * COUT + co) * (size_t)(HH * WW)
                             + (size_t)h * WW + col;
            out[off]          = acc[0][nt][r] + bv;
            out[N1 + off]     = acc[1][nt][r] + bv;
            out[2 * N1 + off] = acc[2][nt][r] + bv;
        }
    }
}

extern "C" void kernel_launch(void* const* d_in, const int* in_sizes, int n_in,
                              void* d_out, int out_size, void* d_ws, size_t ws_size,
                              hipStream_t stream) {
    const float* x    = (const float*)d_in[0];
    const float* xh   = (const float*)d_in[1];
    const float* cw   = (const float*)d_in[2];
    const float* hv   = (const float*)d_in[3];
    const float* bias = (const float*)d_in[4];
    float* out = (float*)d_out;

    const int NW = COUT * CIN * 9;        // 36864
    float* mm  = (float*)d_ws;            // [0]=min, [1]=max
    float* hwt = (float*)d_ws + 16;       // hash_weight, transposed [co][pos][ci]
    float* wt  = hwt + NW;                // conv_weight, transposed [co][pos][ci]

    minmax_kernel<<<1, 256, 0, stream>>>(cw, mm, NW);
    hashw_kernel<<<(NW + 255) / 256, 256, 0, stream>>>(cw, hv, mm, wt, hwt, NW);

    dim3 grid(WW / TILE_W, HH, BB);       // (4, 128, 16)
    conv3_wmma_kernel<<<grid, 128, 0, stream>>>(x, xh, wt, hwt, bias, out);
}